// UninormAggregator_42580305772550
// MI455X (gfx1250) — compile-verified
//
#include <hip/hip_runtime.h>
#include <hip/hip_bf16.h>

typedef __attribute__((ext_vector_type(2))) float v2f;
typedef __attribute__((ext_vector_type(8))) float v8f;

#define NPARAM 8192
#define NIN    4096
#define KSLICES 8
#define KSLICE_LEN (NPARAM / KSLICES)   // 1024

// ---------------------------------------------------------------------------
// pair_uninorm, divides hoisted out (inv_n, inv_1n precomputed per row)
// ---------------------------------------------------------------------------
__device__ __forceinline__ float pair_uni(float a, float b, float n,
                                          float inv_n, float inv_1n) {
    float s   = a + b;
    float y00 = n * fmaxf(0.0f, __builtin_fmaf(s, inv_n, -1.0f));
    float y11 = __builtin_fmaf(1.0f - n,
                               fminf(1.0f, (s - 2.0f * n) * inv_1n), n);
    float yxx = fminf(a, b);
    float y = 0.0f;
    y = (a <= n && b <= n) ? y00 : y;                     // both below/at neutral
    y = (a >= n && b >= n) ? y11 : y;                     // both above/at neutral
    y = ((a > n && b < n) || (a < n && b > n)) ? yxx : y; // strictly opposite
    return y;
}

// ---------------------------------------------------------------------------
// Kernel 1: one wave32 per row. Lane L tree-reduces x[row, L*128 .. L*128+127]
// (adjacent-pair tree preserved), then xor-butterfly across lanes.
// Streams all 134 MB of x exactly once.
// ---------------------------------------------------------------------------
__global__ void uninorm_rows_kernel(const float* __restrict__ x,
                                    const float* __restrict__ neutral,
                                    float* __restrict__ y_out) {
    const int lane = threadIdx.x & 31;
    const int wave = threadIdx.x >> 5;
    const int row  = blockIdx.x * 8 + wave;

    const float n      = neutral[row];
    const float inv_n  = 1.0f / n;
    const float inv_1n = 1.0f / (1.0f - n);

    const float4* p = (const float4*)(x + (size_t)row * NIN + (size_t)lane * 128);

    float v[32];
#pragma unroll
    for (int i = 0; i < 32; ++i) {
        float4 t  = p[i];
        float lo  = pair_uni(t.x, t.y, n, inv_n, inv_1n);
        float hi  = pair_uni(t.z, t.w, n, inv_n, inv_1n);
        v[i] = pair_uni(lo, hi, n, inv_n, inv_1n);
    }
    // in-lane tree: 32 -> 1
#pragma unroll
    for (int len = 16; len >= 1; len >>= 1) {
#pragma unroll
        for (int i = 0; i < len; ++i)
            v[i] = pair_uni(v[2 * i], v[2 * i + 1], n, inv_n, inv_1n);
    }
    // cross-lane tree: 32 lanes -> 1 (pair_uni is symmetric; xor-butterfly
    // reproduces the adjacent-pair tree ordering)
    float r = v[0];
#pragma unroll
    for (int m = 1; m < 32; m <<= 1)
        r = pair_uni(r, __shfl_xor(r, m, 32), n, inv_n, inv_1n);

    if (lane == 0) y_out[row] = r;
}

// ---------------------------------------------------------------------------
// Kernel 2: GEMV partials via V_WMMA_F32_16X16X4_F32.
//   grid = (NPARAM/16/8, KSLICES), block = 256 (8 waves, 1 row-tile each).
//
// K-permutation trick: each lane loads a b128 (8 W-columns across the two
// lane-halves) and feeds TWO wmma ops, halving VMEM instruction count:
//   wmma j in {0,1}, kb = kk + 8u:
//     A lo-lanes  (K0,K1) <- W[row, kb+2j   .. kb+2j+1]
//     A hi-lanes  (K2,K3) <- W[row, kb+2j+4 .. kb+2j+5]
//     B col 0: lane0  <- y[kb+2j .. +1],  lane16 <- y[kb+2j+4 .. +5]
//   D column 0: lane 0 = rows m0..m0+7, lane 16 = rows m0+8..m0+15.
//
// y slice staged into LDS via GLOBAL_LOAD_ASYNC_TO_LDS_B128 (ASYNCcnt DMA).
// ---------------------------------------------------------------------------
__global__ void gemv_wmma_kernel(const float* __restrict__ W,
                                 const float* __restrict__ y,
                                 float* __restrict__ partial) {
    __shared__ float lds_y[KSLICE_LEN];

    const int lane = threadIdx.x & 31;
    const int wave = threadIdx.x >> 5;
    const int s    = blockIdx.y;
    const int tile = blockIdx.x * 8 + wave;
    const int m0   = tile * 16;

    // cooperative async DMA of this K-slice of y into LDS
    // (256 threads x 16 B = 4 KB). LDS offset = low 32 bits of generic ptr
    // (hardware's flat->LDS truncation rule).
    {
        unsigned int ldsa = (unsigned int)(uintptr_t)(&lds_y[threadIdx.x * 4]);
        unsigned long long ga =
            (unsigned long long)(uintptr_t)(y + (size_t)s * KSLICE_LEN
                                              + (size_t)threadIdx.x * 4);
        asm volatile("global_load_async_to_lds_b128 %0, %1, off"
                     :: "v"(ldsa), "v"(ga) : "memory");
        asm volatile("s_wait_asynccnt 0x0" ::: "memory");
    }
    __syncthreads();

    const int  hiK  = lane >> 4;            // 0: K-slots 0,1   1: K-slots 2,3
    const bool bact = (lane & 15) == 0;     // lanes carrying B column 0

    // b128-aligned per-lane W pointer: lo-lanes at +0, hi-lanes at +4 cols
    const float* wrow = W + (size_t)(m0 + (lane & 15)) * NPARAM
                          + (size_t)s * KSLICE_LEN + 4 * hiK;

    v8f acc = {};
    for (int kk = 0; kk < KSLICE_LEN; kk += 16) {
        __builtin_prefetch(wrow + kk + 256, 0, 0);   // global_prefetch_b8
#pragma unroll
        for (int u = 0; u < 2; ++u) {
            const int kb = kk + 8 * u;
            float4 wv = *(const float4*)(wrow + kb);          // 4 cols / lane-half
            v2f a0; a0.x = wv.x; a0.y = wv.y;                 // cols kb+2*0(+4*hiK)
            v2f a1; a1.x = wv.z; a1.y = wv.w;                 // cols kb+2*1(+4*hiK)
            v2f yv0 = *(const v2f*)(&lds_y[kb + 4 * hiK]);
            v2f yv1 = *(const v2f*)(&lds_y[kb + 4 * hiK + 2]);
            v2f b0; b0.x = bact ? yv0.x : 0.0f; b0.y = bact ? yv0.y : 0.0f;
            v2f b1; b1.x = bact ? yv1.x : 0.0f; b1.y = bact ? yv1.y : 0.0f;
            acc = __builtin_amdgcn_wmma_f32_16x16x4_f32(
                false, a0, false, b0, (short)0, acc, false, false);
            acc = __builtin_amdgcn_wmma_f32_16x16x4_f32(
                false, a1, false, b1, (short)0, acc, false, false);
        }
    }

    if (bact) { // lane 0 -> rows m0..m0+7, lane 16 -> rows m0+8..m0+15
        float* dst = partial + (size_t)s * NPARAM + m0 + hiK * 8;
#pragma unroll
        for (int r = 0; r < 8; ++r) dst[r] = acc[r];
    }
}

// ---------------------------------------------------------------------------
// Kernel 3: out[m] = b[m] + sum_s partial[s][m]  (deterministic reduction)
// ---------------------------------------------------------------------------
__global__ void finalize_kernel(const float* __restrict__ partial,
                                const float* __restrict__ b,
                                float* __restrict__ out) {
    const int m = blockIdx.x * blockDim.x + threadIdx.x;
    float acc = b[m];
#pragma unroll
    for (int s = 0; s < KSLICES; ++s)
        acc += partial[(size_t)s * NPARAM + m];
    out[m] = acc;
}

extern "C" void kernel_launch(void* const* d_in, const int* in_sizes, int n_in,
                              void* d_out, int out_size, void* d_ws, size_t ws_size,
                              hipStream_t stream) {
    (void)in_sizes; (void)n_in; (void)out_size; (void)ws_size;

    const float* x       = (const float*)d_in[0];
    const float* neutral = (const float*)d_in[1];
    const float* W       = (const float*)d_in[2];
    const float* b       = (const float*)d_in[3];
    float*       out     = (float*)d_out;

    // workspace layout: y[8192] | partial[KSLICES * 8192]   (~288 KB)
    float* y       = (float*)d_ws;
    float* partial = y + NPARAM;

    uninorm_rows_kernel<<<NPARAM / 8, 256, 0, stream>>>(x, neutral, y);
    gemv_wmma_kernel<<<dim3(NPARAM / 16 / 8, KSLICES), 256, 0, stream>>>(W, y, partial);
    finalize_kernel<<<NPARAM / 256, 256, 0, stream>>>(partial, b, out);
}